// MeshGenLoss_9268539424872
// MI455X (gfx1250) — compile-verified
//
#include <hip/hip_runtime.h>
#include <hip/hip_bf16.h>
#include <float.h>
#include <math.h>

// ---------------------------------------------------------------------------
// MeshGenLoss fused pipeline for MI455X (gfx1250, wave32, WMMA)
//
// Chamfer + density distances as rank-4 GEMMs on the WMMA unit:
//   d(a,b) = |a|^2 + |b|^2 - 2 a.b
//   A row  = (ax, ay, az, |a|^2)          (16x4 f32 A fragment)
//   B col  = (-2bx, -2by, -2bz, 1)        (4x16 f32 B fragment)
//   C      = |b|^2 broadcast per column   (16x16 f32 C fragment)
//   D = A*B + C  ->  exact f32 squared distances, one v_wmma per 16x16 tile.
// Row-minima kept in the 8 accumulator VGPRs, reduced with 16-lane shuffles,
// partials merged with global_atomic_min_num_f32. The 8192x8192 distance
// matrix never touches memory. Columns are chunked 8-way (2048 blocks / 16K
// waves per direction) and the inner loop software-pipelines the next b128
// tile load over the current tile's WMMA. The diagonal mask (density term)
// is a template parameter so the chamfer hot loop carries zero overhead.
// ---------------------------------------------------------------------------

typedef __attribute__((ext_vector_type(2))) float v2f;
typedef __attribute__((ext_vector_type(8))) float v8f;

#define TB 256           // threads per block (8 wave32)
#define WG_ROWS 128      // A-rows per workgroup (8 waves * 16 rows)

constexpr int Bn = 4;      // batch
constexpr int Nn = 8192;   // pred points
constexpr int Mn = 8192;   // target points
constexpr int Ln = 512;    // latent
constexpr int Sn = 1024;   // subsample

// --- zero the scalar accumulators (re-run every launch: graph-replay safe)
__global__ void zero_acc(float* acc) {
    if (threadIdx.x < 8) acc[threadIdx.x] = 0.0f;
}

// --- fill a float buffer with a constant (init min buffers to +FLT_MAX)
__global__ void fill_value(float* __restrict__ p, int n, float v) {
    int i = blockIdx.x * blockDim.x + threadIdx.x;
    if (i < n) p[i] = v;
}

// --- (x,y,z) -> (x,y,z,|p|^2) padded float4 for b128 loads in the hot loop
__global__ void pad_points(const float* __restrict__ src, float4* __restrict__ dst, int count) {
    int i = blockIdx.x * blockDim.x + threadIdx.x;
    if (i < count) {
        float x = src[3 * i + 0], y = src[3 * i + 1], z = src[3 * i + 2];
        dst[i] = make_float4(x, y, z, x * x + y * y + z * z);
    }
}

// --- gather subsampled points, padded
__global__ void gather_points(const float* __restrict__ pos, const int* __restrict__ sub_idx,
                              float4* __restrict__ dst, int nPts, int nSub, int total) {
    int t = blockIdx.x * blockDim.x + threadIdx.x;
    if (t < total) {
        int b = t / nSub;
        int i = t - b * nSub;
        int j = sub_idx[i];
        const float* p = pos + ((size_t)b * nPts + j) * 3;
        float x = p[0], y = p[1], z = p[2];
        dst[t] = make_float4(x, y, z, x * x + y * y + z * z);
    }
}

// --- fused WMMA distance + row-min kernel (column-chunked).
//  Each wave owns a 16-row A tile and scans one column chunk in 16-col tiles,
//  keeping per-slot minima in registers; the final per-row partial min is
//  merged into minOut with a float atomic min (values pre-filled FLT_MAX).
//  DIAG!=0 adds +1e6 on the (within-batch) diagonal (density eye mask);
//  compile-time so the chamfer instantiation's loop is branch-free.
template <int DIAG>
__global__ void wmma_min_rows(const float4* __restrict__ A, const float4* __restrict__ Bp,
                              float* __restrict__ minOut, int nA, int nB,
                              int rowTiles, int colChunks, int colsPerChunk) {
    int cc    = blockIdx.x % colChunks;
    int rem   = blockIdx.x / colChunks;
    int chunk = rem % rowTiles;
    int batch = rem / rowTiles;

    int lane  = threadIdx.x & 31;
    int wave  = threadIdx.x >> 5;
    int half  = lane >> 4;      // 0: lanes 0-15 (K=0,1 / rows 0-7), 1: lanes 16-31 (K=2,3 / rows 8-15)
    int l16   = lane & 15;

    int rowBase = chunk * WG_ROWS + wave * 16;   // within-batch row tile base
    int rA = rowBase + l16;

    // A fragment: 16x4 f32. lanes 0-15: V0=K0(ax) V1=K1(ay); lanes 16-31: V0=K2(az) V1=K3(|a|^2)
    float4 ap = A[(size_t)batch * nA + rA];
    v2f afrag;
    afrag.x = half ? ap.z : ap.x;
    afrag.y = half ? ap.w : ap.y;

    v8f rmin;
#pragma unroll
    for (int v = 0; v < 8; ++v) rmin[v] = FLT_MAX;

    int colBase0 = cc * colsPerChunk;
    const float4* cb = Bp + (size_t)batch * nB + colBase0;
    int ntiles = colsPerChunk >> 4;
    int lastT  = ntiles - 1;

    // Software-pipelined: tile t+1's b128 load is in flight while tile t's
    // fragments + WMMA + min execute.
    float4 bp = cb[l16];
#pragma unroll 2
    for (int ct = 0; ct < ntiles; ++ct) {
        int nt = (ct != lastT) ? (ct + 1) : ct;
        float4 bnx = cb[(nt << 4) + l16];

        // B fragment: 4x16 f32. lanes 0-15: V0=K0 V1=K1; lanes 16-31: V0=K2 V1=K3
        v2f bfrag;
        bfrag.x = half ? (-2.0f * bp.z) : (-2.0f * bp.x);
        bfrag.y = half ? 1.0f           : (-2.0f * bp.y);

        // C fragment: every accumulator slot of this lane is column l16 -> |b|^2
        float csq = bp.w;
        v8f c;
#pragma unroll
        for (int v = 0; v < 8; ++v) c[v] = csq;

        // D[r][c] = |a_r|^2 - 2 a_r.b_c + |b_c|^2
        v8f d = __builtin_amdgcn_wmma_f32_16x16x4_f32(
            /*neg_a=*/false, afrag, /*neg_b=*/false, bfrag,
            /*c_mod=*/(short)0, c, /*reuse_a=*/false, /*reuse_b=*/false);

        if (DIAG) {
            // Diagonal only intersects this wave's 16-row tile in the single
            // column tile where colBase == rowBase; each lane owns at most one
            // diagonal slot (v = l16 - 8*half).
            if ((colBase0 + (ct << 4)) == rowBase) {
                int v = l16 - (half << 3);
                if (v >= 0 && v < 8) d[v] += 1.0e6f;
            }
        }

#pragma unroll
        for (int v = 0; v < 8; ++v) rmin[v] = fminf(rmin[v], d[v]);
        bp = bnx;
    }

    // Per-row partial min: reduce across the 16 lanes of each half, then
    // merge with a global float atomic-min.
#pragma unroll
    for (int v = 0; v < 8; ++v) {
        float m = rmin[v];
        for (int mask = 8; mask >= 1; mask >>= 1)
            m = fminf(m, __shfl_xor(m, mask, 16));
        if (l16 == 0) {
            int row = rowBase + v + (half ? 8 : 0);
            atomicMin(&minOut[(size_t)batch * nA + row], m);
        }
    }
}

// --- grid-stride sum -> atomicAdd into accumulator
__global__ void sum_kernel(const float* __restrict__ x, int n, float* acc) {
    __shared__ float sm[TB];
    int t = blockIdx.x * blockDim.x + threadIdx.x;
    float s = 0.0f;
    for (int i = t; i < n; i += gridDim.x * blockDim.x) s += x[i];
    sm[threadIdx.x] = s;
    __syncthreads();
    for (int o = TB / 2; o > 0; o >>= 1) {
        if (threadIdx.x < o) sm[threadIdx.x] += sm[threadIdx.x + o];
        __syncthreads();
    }
    if (threadIdx.x == 0) atomicAdd(acc, sm[0]);
}

// --- grid-stride squared-error sum
__global__ void mse_kernel(const float* __restrict__ a, const float* __restrict__ b,
                           int n, float* acc) {
    __shared__ float sm[TB];
    int t = blockIdx.x * blockDim.x + threadIdx.x;
    float s = 0.0f;
    for (int i = t; i < n; i += gridDim.x * blockDim.x) {
        float d = a[i] - b[i];
        s += d * d;
    }
    sm[threadIdx.x] = s;
    __syncthreads();
    for (int o = TB / 2; o > 0; o >>= 1) {
        if (threadIdx.x < o) sm[threadIdx.x] += sm[threadIdx.x + o];
        __syncthreads();
    }
    if (threadIdx.x == 0) atomicAdd(acc, sm[0]);
}

// --- sum of (1 + logvar - mu^2 - exp(logvar))
__global__ void kl_kernel(const float* __restrict__ mu, const float* __restrict__ lv,
                          int n, float* acc) {
    __shared__ float sm[TB];
    int t = blockIdx.x * blockDim.x + threadIdx.x;
    float s = 0.0f;
    for (int i = t; i < n; i += gridDim.x * blockDim.x) {
        float m = mu[i], l = lv[i];
        s += 1.0f + l - m * m - expf(l);
    }
    sm[threadIdx.x] = s;
    __syncthreads();
    for (int o = TB / 2; o > 0; o >>= 1) {
        if (threadIdx.x < o) sm[threadIdx.x] += sm[threadIdx.x + o];
        __syncthreads();
    }
    if (threadIdx.x == 0) atomicAdd(acc, sm[0]);
}

// --- single block: density std (ddof=1) per batch + weighted total
__global__ void combine_kernel(const float* __restrict__ nnd, const float* __restrict__ acc,
                               float* __restrict__ out) {
    __shared__ float s1[TB];
    __shared__ float s2[TB];
    __shared__ float densSum;
    int t = threadIdx.x;
    if (t == 0) densSum = 0.0f;
    __syncthreads();
    for (int b = 0; b < Bn; ++b) {
        float s = 0.0f, q = 0.0f;
        for (int i = t; i < Sn; i += TB) {
            float v = nnd[b * Sn + i];
            s += v;
            q += v * v;
        }
        s1[t] = s; s2[t] = q;
        __syncthreads();
        for (int o = TB / 2; o > 0; o >>= 1) {
            if (t < o) { s1[t] += s1[t + o]; s2[t] += s2[t + o]; }
            __syncthreads();
        }
        if (t == 0) {
            float sum = s1[0], sq = s2[0];
            float var = (sq - sum * sum * (1.0f / (float)Sn)) * (1.0f / (float)(Sn - 1));
            densSum += sqrtf(fmaxf(var, 0.0f));
        }
        __syncthreads();
    }
    if (t == 0) {
        float density  = densSum * (1.0f / (float)Bn);
        float cd       = acc[0] * (1.0f / (float)(Bn * Nn));     // sum(minA)+sum(minB), N==M
        float kl       = -0.5f * acc[1] * (1.0f / (float)(Bn * Ln));
        float sizing   = acc[2] * (1.0f / (float)(Bn * Nn));
        float material = acc[3] * (1.0f / (float)(Bn * Nn * 4));
        out[0] = cd + 0.001f * kl + 0.1f * density + 0.05f * sizing + 0.1f * material;
    }
}

extern "C" void kernel_launch(void* const* d_in, const int* in_sizes, int n_in,
                              void* d_out, int out_size, void* d_ws, size_t ws_size,
                              hipStream_t stream) {
    const float* pred_pos        = (const float*)d_in[0];
    const float* pred_sizing     = (const float*)d_in[1];
    const float* pred_material   = (const float*)d_in[2];
    const float* target_pos      = (const float*)d_in[3];
    const float* target_sizing   = (const float*)d_in[4];
    const float* target_material = (const float*)d_in[5];
    const float* mu              = (const float*)d_in[6];
    const float* logvar          = (const float*)d_in[7];
    const int*   sub_idx         = (const int*)d_in[8];
    (void)in_sizes; (void)n_in; (void)out_size; (void)ws_size;

    // Workspace layout (16B-aligned sections; ~1.4 MB total)
    float4* padP = (float4*)d_ws;            // Bn*Nn padded pred points
    float4* padT = padP + (size_t)Bn * Nn;   // Bn*Mn padded target points
    float4* pts4 = padT + (size_t)Bn * Mn;   // Bn*Sn gathered subsample
    float*  minA = (float*)(pts4 + (size_t)Bn * Sn);  // Bn*Nn
    float*  minB = minA + (size_t)Bn * Nn;            // Bn*Mn (contiguous after minA)
    float*  nnd  = minB + (size_t)Bn * Mn;            // Bn*Sn (contiguous after minB)
    float*  acc  = nnd + (size_t)Bn * Sn;             // 8 scalar accumulators

    zero_acc<<<1, 32, 0, stream>>>(acc);

    // Pre-fill all min buffers (minA|minB|nnd contiguous) with +FLT_MAX.
    int nMin = Bn * Nn + Bn * Mn + Bn * Sn;
    fill_value<<<(nMin + TB - 1) / TB, TB, 0, stream>>>(minA, nMin, FLT_MAX);

    pad_points<<<(Bn * Nn + TB - 1) / TB, TB, 0, stream>>>(pred_pos, padP, Bn * Nn);
    pad_points<<<(Bn * Mn + TB - 1) / TB, TB, 0, stream>>>(target_pos, padT, Bn * Mn);
    gather_points<<<(Bn * Sn + TB - 1) / TB, TB, 0, stream>>>(pred_pos, sub_idx, pts4, Nn, Sn, Bn * Sn);

    // Chamfer: pred->target and target->pred (8-way column chunking, 2048 blocks each),
    // plus density self-distances with diagonal mask.
    {
        int rowTiles = Nn / WG_ROWS, colChunks = 8, cpc = Mn / colChunks;
        wmma_min_rows<0><<<Bn * rowTiles * colChunks, TB, 0, stream>>>(
            padP, padT, minA, Nn, Mn, rowTiles, colChunks, cpc);
        wmma_min_rows<0><<<Bn * rowTiles * colChunks, TB, 0, stream>>>(
            padT, padP, minB, Mn, Nn, rowTiles, colChunks, cpc);
    }
    {
        int rowTiles = Sn / WG_ROWS, colChunks = 2, cpc = Sn / colChunks;
        wmma_min_rows<1><<<Bn * rowTiles * colChunks, TB, 0, stream>>>(
            pts4, pts4, nnd, Sn, Sn, rowTiles, colChunks, cpc);
    }

    sum_kernel<<<128, TB, 0, stream>>>(minA, Bn * Nn + Bn * Mn, acc + 0); // minA & minB contiguous
    kl_kernel<<<8, TB, 0, stream>>>(mu, logvar, Bn * Ln, acc + 1);
    mse_kernel<<<64, TB, 0, stream>>>(pred_sizing, target_sizing, Bn * Nn, acc + 2);
    mse_kernel<<<128, TB, 0, stream>>>(pred_material, target_material, Bn * Nn * 4, acc + 3);

    combine_kernel<<<1, TB, 0, stream>>>(nnd, acc, (float*)d_out);
}